// LastMessageAggregator_72052371357813
// MI455X (gfx1250) — compile-verified
//
#include <hip/hip_runtime.h>

// 128-bit vector type for b128 loads/stores.
typedef float float4v __attribute__((ext_vector_type(4)));

// ---------------------------------------------------------------------------
// Kernel 1: last_pos[n] = -1 for all nodes (workspace must be re-inited every
// call; the harness does not re-poison between graph replays).
// ---------------------------------------------------------------------------
__global__ void lma_init(int* __restrict__ last_pos, int n_nodes) {
    int i = blockIdx.x * blockDim.x + threadIdx.x;
    if (i < n_nodes) last_pos[i] = -1;
}

// ---------------------------------------------------------------------------
// Kernel 2: scatter-max of event position into last_pos[node_ids[i]].
// atomicMax with unused return lowers to no-return global_atomic_max_i32;
// the 400KB last_pos array is L2-resident so the RMWs stay on-chip.
// ---------------------------------------------------------------------------
__global__ void lma_scatter_max(const int* __restrict__ node_ids,
                                int* __restrict__ last_pos, int b) {
    int i = blockIdx.x * blockDim.x + threadIdx.x;
    if (i < b) {
        atomicMax(last_pos + node_ids[i], i);
    }
}

// ---------------------------------------------------------------------------
// Kernel 3: per-node gather.
// Each group of (D/4) lanes owns one node row; each lane moves one float4
// (global_load_b128 / global_store_b128). Non-temporal hints on both sides:
// every gathered row and every output element is touched exactly once.
// `shift` = log2(D/4) so the node/lane split is shifts+masks, not int div.
// ---------------------------------------------------------------------------
__global__ void lma_gather(const float* __restrict__ messages,
                           const float* __restrict__ timestamps,
                           const int* __restrict__ last_pos,
                           float* __restrict__ out_msg,
                           float* __restrict__ out_ts,
                           float* __restrict__ out_mask,
                           int n_nodes, int d, int shift) {
    const int vecs = 1 << shift;                      // float4s per row (64)
    const int node = blockIdx.x * (blockDim.x >> shift) + (threadIdx.x >> shift);
    const int lane = threadIdx.x & (vecs - 1);
    if (node >= n_nodes) return;

    const int lp = last_pos[node];                    // uniform per 64-lane group

    float4v v = {0.0f, 0.0f, 0.0f, 0.0f};
    if (lp >= 0) {
        const float4v* src =
            reinterpret_cast<const float4v*>(messages + (size_t)lp * (size_t)d);
        v = __builtin_nontemporal_load(src + lane);   // NT global_load_b128
    }
    float4v* dst = reinterpret_cast<float4v*>(out_msg + (size_t)node * (size_t)d);
    __builtin_nontemporal_store(v, dst + lane);       // NT global_store_b128

    if (lane == 0) {
        float ts = 0.0f, m = 0.0f;
        if (lp >= 0) { ts = timestamps[lp]; m = 1.0f; }
        __builtin_nontemporal_store(ts, out_ts + node);
        __builtin_nontemporal_store(m, out_mask + node);
    }
}

// ---------------------------------------------------------------------------
// Launcher. Inputs (setup_inputs order): node_ids[B] i32, messages[B*D] f32,
// timestamps[B] f32, n_nodes scalar (device-resident; NOT read on host —
// N is derived from out_size = N*(D+2) to stay graph-capture safe).
// d_out layout (tuple concat): [N*D messages | N timestamps | N mask].
// ---------------------------------------------------------------------------
extern "C" void kernel_launch(void* const* d_in, const int* in_sizes, int n_in,
                              void* d_out, int out_size, void* d_ws, size_t ws_size,
                              hipStream_t stream) {
    const int*   node_ids   = (const int*)d_in[0];
    const float* messages   = (const float*)d_in[1];
    const float* timestamps = (const float*)d_in[2];

    const int b = in_sizes[0];
    const int d = in_sizes[1] / in_sizes[0];      // 256
    const int n_nodes = out_size / (d + 2);       // 100000

    int* last_pos = (int*)d_ws;                   // N * 4 bytes of scratch

    float* out_msg  = (float*)d_out;
    float* out_ts   = out_msg + (size_t)n_nodes * (size_t)d;
    float* out_mask = out_ts + n_nodes;

    // Phase 1a: init scatter target.
    lma_init<<<(n_nodes + 255) / 256, 256, 0, stream>>>(last_pos, n_nodes);

    // Phase 1b: scatter-max of event positions (1M L2-resident atomics).
    lma_scatter_max<<<(b + 255) / 256, 256, 0, stream>>>(node_ids, last_pos, b);

    // Phase 2: gather rows + write ts/mask. 64 lanes per node, 4 nodes per
    // 256-thread block (8 wave32s per block).
    const int vecs = d / 4;                       // 64 float4s per row
    int shift = 0;
    while ((1 << shift) < vecs) ++shift;          // log2(64) = 6
    const int nodes_per_block = 256 >> shift;     // 4
    const int grid = (n_nodes + nodes_per_block - 1) / nodes_per_block;
    lma_gather<<<grid, 256, 0, stream>>>(messages, timestamps, last_pos,
                                         out_msg, out_ts, out_mask,
                                         n_nodes, d, shift);
}